// NonLocalBlock_37288906064082
// MI455X (gfx1250) — compile-verified
//
#include <hip/hip_runtime.h>
#include <hip/hip_bf16.h>

// NonLocalBlock, algebraically fused for MI455X (gfx1250, wave32, WMMA).
// B=4, C=128, N=4096, K=32.

typedef __attribute__((ext_vector_type(2))) float v2f;
typedef __attribute__((ext_vector_type(8))) float v8f;

#define NB_B 4
#define NB_N 4096
#define NB_C 128
#define NB_K 32
#define NB_R (NB_B * NB_N)   // 16384 rows (b,n)

// ---------------------------------------------------------------------------
// K0: tiny weight-fusion precompute.
//   G[c][e]  = sum_d Wt[d][c]*Wt[d][e]         (Wt^T Wt)
//   h[e]     = sum_d bt[d]*Wt[d][e]
//   Qt[c][e] = sum_d Ww[e][d]*Wg[d][c]         ((Ww*Wg) transposed)
//   r[e]     = sum_d Ww[e][d]*bg[d] + bw[e]
// grid: 128 blocks x 128 threads; block index i, thread index j.
// ---------------------------------------------------------------------------
__global__ __launch_bounds__(128) void precompute_kernel(
    const float* __restrict__ Wt, const float* __restrict__ bt,
    const float* __restrict__ Wg, const float* __restrict__ bg,
    const float* __restrict__ Ww, const float* __restrict__ bw,
    float* __restrict__ G, float* __restrict__ Qt,
    float* __restrict__ h, float* __restrict__ r) {
  const int i = blockIdx.x;   // c for G, e for Qt/r
  const int j = threadIdx.x;  // e for G/h, c for Qt
  float g = 0.f, q = 0.f, hv = 0.f, rv = 0.f;
  for (int d = 0; d < NB_C; ++d) {
    g  = fmaf(Wt[d * NB_C + i], Wt[d * NB_C + j], g);
    q  = fmaf(Ww[i * NB_C + d], Wg[d * NB_C + j], q);
    hv = fmaf(bt[d], Wt[d * NB_C + j], hv);
    rv = fmaf(Ww[i * NB_C + d], bg[d], rv);
  }
  G[i * NB_C + j]  = g;        // G[c][e]
  Qt[j * NB_C + i] = q;        // Qt[c][e] = (Ww*Wg)[e][c]
  if (i == 0) h[j] = hv;
  if (j == 0) r[i] = rv + bw[i];
}

__global__ void zero_kernel(float* __restrict__ p, int n) {
  int i = blockIdx.x * blockDim.x + threadIdx.x;
  if (i < n) p[i] = 0.f;
}

// ---------------------------------------------------------------------------
// Shared WMMA f32 16x16x4 GEMM tile body.
// Computes acc(16x16) = Xs(16x128) * Bm(128x16) where Bm[k][n] = Bsrc[k*128+e0+n].
// A fragment (ISA 7.12.2, 32-bit A 16x4): V0 = K{0|2}, V1 = K{1|3}, split at lane 16.
// B fragment: V0 = row K{0|2} striped over lanes (N = lane&15), V1 = K{1|3}.
// C/D: VGPR v -> M = v + 8*(lane>=16), N = lane&15.
// ---------------------------------------------------------------------------
__device__ __forceinline__ v8f wmma_tile_f32(const float Xs[16][132],
                                             const float* __restrict__ Bsrc,
                                             int e /* = e0 + (lane&15) */,
                                             int hi /* lane>=16 */,
                                             int m /* lane&15 */) {
  v8f acc = {};
#pragma unroll 4
  for (int k = 0; k < NB_C; k += 4) {
    v2f a, b;
    a.x = Xs[m][k + 2 * hi];
    a.y = Xs[m][k + 2 * hi + 1];
    b.x = Bsrc[(k + 2 * hi) * NB_C + e];
    b.y = Bsrc[(k + 2 * hi + 1) * NB_C + e];
    acc = __builtin_amdgcn_wmma_f32_16x16x4_f32(false, a, false, b,
                                                (short)0, acc, false, false);
  }
  return acc;
}

__device__ __forceinline__ void load_x_tile(float Xs[16][132],
                                            const float* __restrict__ X,
                                            int row0, int tid) {
  for (int i = tid; i < 16 * NB_C; i += 256) {
    int rr = i >> 7, cc = i & 127;
    Xs[rr][cc] = X[(size_t)(row0 + rr) * NB_C + cc];
  }
}

// ---------------------------------------------------------------------------
// K1: t2 = xr * G + h.   xr viewed [16384,128] (pure reshape of x).
// block = 256 thr (8 waves) -> 16 rows x 128 cols; grid = 1024.
// ---------------------------------------------------------------------------
__global__ __launch_bounds__(256) void gemm_t2_kernel(
    const float* __restrict__ X, const float* __restrict__ G,
    const float* __restrict__ h, float* __restrict__ T2) {
  __shared__ float Xs[16][132];  // pad 132: lane m hits bank 4m+k -> no conflict
  const int tid = threadIdx.x;
  const int row0 = blockIdx.x * 16;
  load_x_tile(Xs, X, row0, tid);
  __syncthreads();
  const int lane = tid & 31, wave = tid >> 5;
  const int hi = lane >> 4, m = lane & 15;
  const int e = wave * 16 + m;
  v8f acc = wmma_tile_f32(Xs, G, e, hi, m);
  const float bias = h[e];
  float* out = T2 + (size_t)(row0 + 8 * hi) * NB_C + e;
#pragma unroll
  for (int v = 0; v < 8; ++v) out[(size_t)v * NB_C] = acc[v] + bias;
}

// ---------------------------------------------------------------------------
// K2: logits + softmax.  One wave per (b,n); lane = neighbor k.
// logits[k] = sum_c t2[b,n,c] * xr[b, idx[k], c]   (softmax-shift constant dropped)
// ---------------------------------------------------------------------------
__global__ __launch_bounds__(256) void attn_kernel(
    const int* __restrict__ gidx, const float* __restrict__ X,
    const float* __restrict__ T2, float* __restrict__ coef) {
  const int wave = threadIdx.x >> 5, lane = threadIdx.x & 31;
  const size_t bn = (size_t)blockIdx.x * 8 + wave;      // 0..16383
  const int b = (int)(bn >> 12);                        // N = 4096
  const int idx = gidx[bn * NB_K + lane] & (NB_N - 1);
  const float4* xrow = (const float4*)(X + ((size_t)b * NB_N + idx) * NB_C);
  const float4* trow = (const float4*)(T2 + bn * NB_C);
  float acc = 0.f;
#pragma unroll 8
  for (int j = 0; j < NB_C / 4; ++j) {
    float4 t = trow[j];
    float4 xv = xrow[j];
    acc = fmaf(t.x, xv.x, acc); acc = fmaf(t.y, xv.y, acc);
    acc = fmaf(t.z, xv.z, acc); acc = fmaf(t.w, xv.w, acc);
  }
  float mx = acc;
#pragma unroll
  for (int off = 16; off > 0; off >>= 1) mx = fmaxf(mx, __shfl_xor(mx, off, 32));
  const float ex = __expf(acc - mx);
  float sm = ex;
#pragma unroll
  for (int off = 16; off > 0; off >>= 1) sm += __shfl_xor(sm, off, 32);
  coef[bn * NB_K + lane] = ex / sm;
}

// ---------------------------------------------------------------------------
// K3: agg[b,n,c] = sum_k coef[b,n,k] * feature[b,c,n,k].  The only HBM-heavy
// pass (268 MB of feature); each thread streams 128 contiguous bytes (float4).
// block = 256 thr handles (b, 8 n's, all 128 c); grid = 2048.
// ---------------------------------------------------------------------------
__global__ __launch_bounds__(256) void agg_kernel(
    const float* __restrict__ F, const float* __restrict__ coef,
    float* __restrict__ agg) {
  __shared__ float cs[8][NB_K];
  const int tid = threadIdx.x;
  const int b = blockIdx.x >> 9;              // 512 n-tiles per batch
  const int n0 = (blockIdx.x & 511) * 8;
  {
    int nl = tid >> 5, k = tid & 31;
    cs[nl][k] = coef[((size_t)b * NB_N + n0 + nl) * NB_K + k];
  }
  __syncthreads();
  const int c = tid & 127, nh = tid >> 7;
#pragma unroll
  for (int it = 0; it < 4; ++it) {
    const int nl = it * 2 + nh;
    const float4* f4 = (const float4*)(
        F + (((size_t)b * NB_C + c) * NB_N + (n0 + nl)) * NB_K);
    float a = 0.f;
#pragma unroll
    for (int j = 0; j < NB_K / 4; ++j) {
      float4 fv = f4[j];
      a = fmaf(fv.x, cs[nl][4 * j + 0], a);
      a = fmaf(fv.y, cs[nl][4 * j + 1], a);
      a = fmaf(fv.z, cs[nl][4 * j + 2], a);
      a = fmaf(fv.w, cs[nl][4 * j + 3], a);
    }
    agg[((size_t)b * NB_N + n0 + nl) * NB_C + c] = a;
  }
}

// ---------------------------------------------------------------------------
// K4: y[b,n,e] = sum_c agg[b,n,c]*Qt[c][e] + r[e]  (row-major [16384,128]),
// fused with BatchNorm batch statistics (sum / sum-of-squares per channel e).
// ---------------------------------------------------------------------------
__global__ __launch_bounds__(256) void gemm_y_kernel(
    const float* __restrict__ A, const float* __restrict__ Qt,
    const float* __restrict__ r, float* __restrict__ Y,
    float* __restrict__ bnSum, float* __restrict__ bnSqr) {
  __shared__ float Xs[16][132];
  const int tid = threadIdx.x;
  const int row0 = blockIdx.x * 16;
  load_x_tile(Xs, A, row0, tid);
  __syncthreads();
  const int lane = tid & 31, wave = tid >> 5;
  const int hi = lane >> 4, m = lane & 15;
  const int e = wave * 16 + m;
  v8f acc = wmma_tile_f32(Xs, Qt, e, hi, m);
  const float bias = r[e];
  float s = 0.f, q = 0.f;
  float* out = Y + (size_t)(row0 + 8 * hi) * NB_C + e;
#pragma unroll
  for (int v = 0; v < 8; ++v) {
    float val = acc[v] + bias;
    out[(size_t)v * NB_C] = val;
    s += val;
    q = fmaf(val, val, q);
  }
  s += __shfl_xor(s, 16, 32);   // combine M=0..7 with M=8..15 halves
  q += __shfl_xor(q, 16, 32);
  if (hi == 0) {
    atomicAdd(&bnSum[e], s);
    atomicAdd(&bnSqr[e], q);
  }
}

// ---------------------------------------------------------------------------
// K5: BatchNorm normalize + transpose [B,N,C] -> [B,C,N] via LDS 32x32 tile.
// grid = (N/32, C/32, B), block = 256.
// ---------------------------------------------------------------------------
__global__ __launch_bounds__(256) void bn_kernel(
    const float* __restrict__ Y, const float* __restrict__ bnSum,
    const float* __restrict__ bnSqr, const float* __restrict__ gamma,
    const float* __restrict__ beta, float* __restrict__ Z) {
  __shared__ float tile[32][33];
  const int n0 = blockIdx.x * 32, e0 = blockIdx.y * 32, b = blockIdx.z;
  const int tj = threadIdx.x & 31, ti = threadIdx.x >> 5;
#pragma unroll
  for (int t = 0; t < 4; ++t) {
    int i = ti + 8 * t;  // n offset; read coalesced in e
    tile[i][tj] = Y[((size_t)b * NB_N + n0 + i) * NB_C + e0 + tj];
  }
  __syncthreads();
  const float inv_cnt = 1.0f / (float)NB_R;
#pragma unroll
  for (int t = 0; t < 4; ++t) {
    int i = ti + 8 * t;  // e offset; write coalesced in n
    int e = e0 + i;
    float mean = bnSum[e] * inv_cnt;
    float var = bnSqr[e] * inv_cnt - mean * mean;
    float scale = gamma[e] * rsqrtf(var + 1e-5f);
    float shift = beta[e] - mean * scale;
    Z[((size_t)b * NB_C + e) * NB_N + n0 + tj] = fmaf(tile[tj][i], scale, shift);
  }
}

// ---------------------------------------------------------------------------
extern "C" void kernel_launch(void* const* d_in, const int* in_sizes, int n_in,
                              void* d_out, int out_size, void* d_ws, size_t ws_size,
                              hipStream_t stream) {
  const int*   gidx    = (const int*)d_in[0];
  const float* x       = (const float*)d_in[1];
  const float* feature = (const float*)d_in[2];
  const float* Wt      = (const float*)d_in[3];
  const float* bt      = (const float*)d_in[4];
  const float* Wg      = (const float*)d_in[5];
  const float* bg      = (const float*)d_in[6];
  const float* Ww      = (const float*)d_in[7];
  const float* bw      = (const float*)d_in[8];
  const float* gamma   = (const float*)d_in[9];
  const float* beta    = (const float*)d_in[10];
  float* out = (float*)d_out;

  float* ws   = (float*)d_ws;
  float* t2   = ws;                                  // 16384*128
  float* coef = t2 + (size_t)NB_R * NB_C;            // 16384*32
  float* agg  = coef + (size_t)NB_R * NB_K;          // 16384*128
  float* G    = agg + (size_t)NB_R * NB_C;           // 128*128
  float* Qt   = G + NB_C * NB_C;                     // 128*128
  float* h    = Qt + NB_C * NB_C;                    // 128
  float* r    = h + NB_C;                            // 128
  float* bnS  = r + NB_C;                            // 128
  // bnQ = bnS + 128 (zeroed together with bnS)
  float* bnQ  = bnS + NB_C;
  float* Y    = t2;                                  // reuse: t2 dead after K2

  zero_kernel<<<1, 256, 0, stream>>>(bnS, 2 * NB_C);
  precompute_kernel<<<NB_C, NB_C, 0, stream>>>(Wt, bt, Wg, bg, Ww, bw, G, Qt, h, r);
  gemm_t2_kernel<<<NB_R / 16, 256, 0, stream>>>(x, G, h, t2);
  attn_kernel<<<NB_R / 8, 256, 0, stream>>>(gidx, x, t2, coef);
  agg_kernel<<<NB_B * (NB_N / 8), 256, 0, stream>>>(feature, coef, agg);
  gemm_y_kernel<<<NB_R / 16, 256, 0, stream>>>(agg, Qt, r, Y, bnS, bnQ);
  bn_kernel<<<dim3(NB_N / 32, NB_C / 32, NB_B), 256, 0, stream>>>(
      Y, bnS, bnQ, gamma, beta, out);
}